// SynchronizationBlock_56040733278598
// MI455X (gfx1250) — compile-verified
//
#include <hip/hip_runtime.h>
#include <math.h>

// ---------------------------------------------------------------------------
// Problem constants (match reference)
// ---------------------------------------------------------------------------
constexpr int   BATCH  = 32;
constexpr int   NDIM   = 2048;
constexpr float LAMBDA = 1.5f;
constexpr float L2C    = LAMBDA * LAMBDA;   // 2.25

// Tiling: 4 waves / 128 threads per block, each wave owns 16 rows.
constexpr int KC    = 32;   // K-chunk per TDM tile
constexpr int TSTR  = 36;   // LDS row stride in floats (32 + 4 pad): 36r mod 64
                            // is a perfect 64-bank permutation for b64 reads
constexpr int WAVES = 4;

typedef float v2f __attribute__((ext_vector_type(2)));
typedef float v8f __attribute__((ext_vector_type(8)));

typedef unsigned int u32x4 __attribute__((ext_vector_type(4)));
typedef int          i32x8 __attribute__((ext_vector_type(8)));
typedef int          i32x4 __attribute__((ext_vector_type(4)));

#if defined(__has_builtin)
#if __has_builtin(__builtin_amdgcn_tensor_load_to_lds)
#define USE_TDM 1
#endif
#endif

// ---------------------------------------------------------------------------
// Kernel 1: scale[b] = -L^2 * (1 - mean_k(x_r^2 + x_i^2))
// ---------------------------------------------------------------------------
__global__ __launch_bounds__(256) void scale_kernel(const float* __restrict__ xr,
                                                    const float* __restrict__ xi,
                                                    float* __restrict__ scale) {
  const int b = blockIdx.x;
  float s = 0.0f;
  for (int i = threadIdx.x; i < NDIM; i += 256) {
    const float r = xr[(size_t)b * NDIM + i];
    const float m = xi[(size_t)b * NDIM + i];
    s = fmaf(r, r, s);
    s = fmaf(m, m, s);
  }
  __shared__ float red[256];
  red[threadIdx.x] = s;
  __syncthreads();
  for (int off = 128; off > 0; off >>= 1) {
    if (threadIdx.x < off) red[threadIdx.x] += red[threadIdx.x + off];
    __syncthreads();
  }
  if (threadIdx.x == 0)
    scale[b] = -L2C * (1.0f - red[0] / (float)NDIM);
}

#ifdef USE_TDM
// One 2D TDM descriptor: tile = 32 (K, dim0) x 16 (rows, dim1) f32,
// tensor_dim0_stride = NDIM, LDS pad: +4 dwords after every 32 dwords
// (-> row stride TSTR=36 floats in LDS).
__device__ __forceinline__ void tdm_load_tile(const float* gptr, unsigned lds_byte) {
  const unsigned long long ga = (unsigned long long)(uintptr_t)gptr;
  u32x4 g0;
  g0.x = 1u;                                            // count=1 (valid), user mode
  g0.y = lds_byte;                                      // lds_addr
  g0.z = (unsigned)(ga & 0xffffffffu);                  // global_addr[31:0]
  g0.w = (unsigned)((ga >> 32) & 0x01ffffffu)           // global_addr[56:32]
         | 0x80000000u;                                 // type=2 ("image")
  i32x8 g1;
  g1[0] = (2 << 16)      // data_size = 4 bytes
        | (1 << 20)      // pad_enable
        | (4 << 22)      // pad_interval: 32 dwords
        | (3 << 25);     // pad_amount: 4 dwords
  g1[1] = (int)((unsigned)NDIM << 16);   // tensor_dim0[15:0] (=2048) in [31:16]
  g1[2] = (int)((unsigned)NDIM << 16);   // tensor_dim0 hi=0 | tensor_dim1[15:0]
  g1[3] = (int)(32u << 16);              // tensor_dim1 hi=0 | tile_dim0=32
  g1[4] = 16;                            // tile_dim1=16, tile_dim2=0
  g1[5] = NDIM;                          // tensor_dim0_stride lo32
  g1[6] = 0;
  g1[7] = 0;
  const i32x4 z4 = {0, 0, 0, 0};
  const i32x8 z8 = {0, 0, 0, 0, 0, 0, 0, 0};
  // clang-23 / therock-10.0 arity: (g0, g1, g2, g3, extra, cpol)
  __builtin_amdgcn_tensor_load_to_lds(g0, g1, z4, z4, z8, 0);
}
#endif

// ---------------------------------------------------------------------------
// Kernel 2: WMMA complex GEMV + epilogue
//   A (SRC0) = x-matrix 16x4: row0 = x_r chunk, row1 = x_i chunk, rows 2..15 = 0
//   B (SRC1) = Y tile 4x16:  B[k][n] = Y[m0+n][k]
//   C_r row0 = Yr@xr (a), row1 = Yr@xi (c)   (row r of D -> VGPR r, lanes 0..15)
//   C_i row0 = Yi@xr (d), row1 = Yi@xi (bb)
//   x1_r = L(a - bb), x1_i = L(c + d)
// ---------------------------------------------------------------------------
__global__ __launch_bounds__(128) void sync_block_kernel(
    const float* __restrict__ Y_r, const float* __restrict__ Y_i,
    const float* __restrict__ x_r, const float* __restrict__ x_i,
    const float* __restrict__ xp_r, const float* __restrict__ xp_i,
    const float* __restrict__ d3_w, const float* __restrict__ d3_b,
    const float* __restrict__ w1, const float* __restrict__ b1,
    const float* __restrict__ w2, const float* __restrict__ b2,
    const float* __restrict__ scale, float* __restrict__ out) {

  __shared__ float s_xr[NDIM];
  __shared__ float s_xi[NDIM];
  __shared__ float s_w1[256];
  __shared__ float s_b1[256];
  __shared__ float s_w2[256];
  __shared__ float s_zero[32];                       // zero pad for A rows 2..15
  __shared__ float s_tile[WAVES][2][2][16 * TSTR];   // [wave][buf][Yr/Yi][...]

  const int b    = blockIdx.y;
  const int tid  = threadIdx.x;
  const int lane = tid & 31;
  const int wave = __builtin_amdgcn_readfirstlane((int)(tid >> 5));

  // ---- block-wide staging of x, MLP weights, zero pad ----
  for (int i = tid; i < NDIM; i += 128) {
    s_xr[i] = x_r[(size_t)b * NDIM + i];
    s_xi[i] = x_i[(size_t)b * NDIM + i];
  }
  for (int i = tid; i < 256; i += 128) {
    s_w1[i] = w1[i];
    s_b1[i] = b1[i];
    s_w2[i] = w2[i];
  }
  if (tid < 32) s_zero[tid] = 0.0f;
  __syncthreads();

  const int m0 = blockIdx.x * (WAVES * 16) + wave * 16;
  const float* __restrict__ Yr = Y_r + ((size_t)b * NDIM + m0) * NDIM;
  const float* __restrict__ Yi = Y_i + ((size_t)b * NDIM + m0) * NDIM;

  const int half = lane >> 4;        // fragment K-half: 0 -> K=0..1, 1 -> K=2..3
  const int n    = lane & 15;        // A row index / B,C,D column index
  const int koff = 2 * half;

  // Branchless per-lane A-operand (x) streaming pointer:
  //   lane row 0 walks s_xr, row 1 walks s_xi, rows 2..15 pin to s_zero.
  // In-chunk step advance (16B) is an immediate DS offset; the pointer is
  // bumped once per chunk by a per-lane amount (KC floats or 0).
  const float* xp     = ((n == 0) ? s_xr : (n == 1) ? s_xi : s_zero) + koff;
  const int    xchunk = (n < 2) ? KC : 0;   // floats per chunk

  v8f cR = {};
  v8f cI = {};

  // 8 WMMA K-steps over one staged chunk. Direct GEPs into s_tile (literal
  // buffer index) so the fragment reads stay in addrspace(3) -> ds_load_b64.
  auto run_steps = [&](const float* tR, const float* tI) {
#pragma unroll
    for (int st = 0; st < KC / 4; ++st) {
      const v2f aF = *(const v2f*)(xp + st * 4);                    // x (A)
      const v2f bR = *(const v2f*)(tR + n * TSTR + st * 4 + koff);  // Yr (B)
      const v2f bI = *(const v2f*)(tI + n * TSTR + st * 4 + koff);  // Yi (B)
      cR = __builtin_amdgcn_wmma_f32_16x16x4_f32(false, aF, false, bR,
                                                 (short)0, cR, false, false);
      cI = __builtin_amdgcn_wmma_f32_16x16x4_f32(false, aF, false, bI,
                                                 (short)0, cI, false, false);
    }
    xp += xchunk;
  };

#ifdef USE_TDM
  // Software pipeline invariant at loop top:
  //   buf0 <- chunk kc      (2 TDM ops in flight or landed)
  //   buf1 <- chunk kc+KC   (2 TDM ops in flight or landed)
  tdm_load_tile(Yr,      (unsigned)(uintptr_t)&s_tile[wave][0][0][0]);
  tdm_load_tile(Yi,      (unsigned)(uintptr_t)&s_tile[wave][0][1][0]);
  tdm_load_tile(Yr + KC, (unsigned)(uintptr_t)&s_tile[wave][1][0][0]);
  tdm_load_tile(Yi + KC, (unsigned)(uintptr_t)&s_tile[wave][1][1][0]);

  // Steady state: branch-free body; final two chunks peeled below.
#pragma unroll 1
  for (int kc = 0; kc < NDIM - 2 * KC; kc += 2 * KC) {
    // TDM ops complete in order per wave: cnt<=2 => chunk kc (oldest 2) landed
    __builtin_amdgcn_s_wait_tensorcnt(2);
    asm volatile("" ::: "memory");
    run_steps(&s_tile[wave][0][0][0], &s_tile[wave][0][1][0]);
    asm volatile("" ::: "memory");
    tdm_load_tile(Yr + kc + 2 * KC, (unsigned)(uintptr_t)&s_tile[wave][0][0][0]);
    tdm_load_tile(Yi + kc + 2 * KC, (unsigned)(uintptr_t)&s_tile[wave][0][1][0]);

    __builtin_amdgcn_s_wait_tensorcnt(2);            // chunk kc+KC landed
    asm volatile("" ::: "memory");
    run_steps(&s_tile[wave][1][0][0], &s_tile[wave][1][1][0]);
    asm volatile("" ::: "memory");
    tdm_load_tile(Yr + kc + 3 * KC, (unsigned)(uintptr_t)&s_tile[wave][1][0][0]);
    tdm_load_tile(Yi + kc + 3 * KC, (unsigned)(uintptr_t)&s_tile[wave][1][1][0]);
  }

  // Tail: chunks NDIM-2*KC (buf0) and NDIM-KC (buf1) already in flight.
  __builtin_amdgcn_s_wait_tensorcnt(2);
  asm volatile("" ::: "memory");
  run_steps(&s_tile[wave][0][0][0], &s_tile[wave][0][1][0]);
  __builtin_amdgcn_s_wait_tensorcnt(0);
  asm volatile("" ::: "memory");
  run_steps(&s_tile[wave][1][0][0], &s_tile[wave][1][1][0]);
#else
  // fallback: synchronous per-wave staging into buffer 0
  for (int kc = 0; kc < NDIM; kc += KC) {
    const int lrow = lane >> 3;
    const int lcol = (lane & 7) * 4;
#pragma unroll
    for (int g = 0; g < 4; ++g) {
      const int row = g * 4 + lrow;
      const float4 vr = *(const float4*)(Yr + (size_t)row * NDIM + kc + lcol);
      const float4 vi = *(const float4*)(Yi + (size_t)row * NDIM + kc + lcol);
      *(float4*)(&s_tile[wave][0][0][0] + row * TSTR + lcol) = vr;
      *(float4*)(&s_tile[wave][0][1][0] + row * TSTR + lcol) = vi;
    }
    run_steps(&s_tile[wave][0][0][0], &s_tile[wave][0][1][0]);
  }
#endif

  // ---- epilogue: lane n (0..15) directly holds row n's four dot products ----
  if (lane < 16) {
    const int   m  = m0 + n;
    const float a  = cR[0];   // Yr@xr
    const float c  = cR[1];   // Yr@xi
    const float d  = cI[0];   // Yi@xr
    const float bb = cI[1];   // Yi@xi

    const float d3w = d3_w[0];
    const float d3b = d3_b[0];
    const float b2v = b2[0];
    const float sb  = scale[b];

    float x1r = LAMBDA * (a - bb);
    float x1i = LAMBDA * (c + d);
    x1r = fmaf(x1r, d3w, d3b);
    x1i = fmaf(x1i, d3w, d3b);
    x1r = fmaf(sb, xp_r[(size_t)b * NDIM + m], x1r);
    x1i = fmaf(sb, xp_i[(size_t)b * NDIM + m], x1i);

    const float ab    = sqrtf(x1r * x1r + x1i * x1i);
    const float denom = fmaxf(ab, 1e-12f);

    float acc = 0.0f;
#pragma unroll 8
    for (int j = 0; j < 256; ++j) {
      const float h = fmaxf(fmaf(ab, s_w1[j], s_b1[j]), 0.0f);
      acc = fmaf(h, s_w2[j], acc);
    }
    const float g = tanhf(acc + b2v);
    const float q = g / denom;

    out[(size_t)b * NDIM + m]                        = x1r * q;   // x1_r * q
    out[(size_t)BATCH * NDIM + (size_t)b * NDIM + m] = x1i * q;   // x1_i * q
  }
}

// ---------------------------------------------------------------------------
// Launcher
// ---------------------------------------------------------------------------
extern "C" void kernel_launch(void* const* d_in, const int* in_sizes, int n_in,
                              void* d_out, int out_size, void* d_ws, size_t ws_size,
                              hipStream_t stream) {
  const float* Y_r  = (const float*)d_in[0];
  const float* Y_i  = (const float*)d_in[1];
  const float* x_r  = (const float*)d_in[2];
  const float* x_i  = (const float*)d_in[3];
  const float* xp_r = (const float*)d_in[4];
  const float* xp_i = (const float*)d_in[5];
  const float* d3_w = (const float*)d_in[6];
  const float* d3_b = (const float*)d_in[7];
  const float* w1   = (const float*)d_in[8];
  const float* b1   = (const float*)d_in[9];
  const float* w2   = (const float*)d_in[10];
  const float* b2   = (const float*)d_in[11];

  float* scale = (float*)d_ws;          // 32 floats of scratch
  float* out   = (float*)d_out;         // [2, B, N] flat

  scale_kernel<<<BATCH, 256, 0, stream>>>(x_r, x_i, scale);

  dim3 grid(NDIM / (WAVES * 16), BATCH);   // (32, 32)
  sync_block_kernel<<<grid, 128, 0, stream>>>(Y_r, Y_i, x_r, x_i, xp_r, xp_i,
                                              d3_w, d3_b, w1, b1, w2, b2,
                                              scale, out);
}